// spectralModel_54941221650713
// MI455X (gfx1250) — compile-verified
//
#include <hip/hip_runtime.h>
#include <math.h>

// ---------------- problem constants ----------------
constexpr int   B    = 2;
constexpr int   C    = 64;
constexpr int   H    = 48;
constexpr int   W    = 48;
constexpr int   WF   = 25;            // W/2 + 1
constexpr int   P    = H * WF;        // 1200 spectral points (= 75 * 16, exact WMMA tiling)
constexpr int   NH   = 8;
constexpr int   HD   = C / NH;        // 8
constexpr int   NREAL = B * C * H * W;   // 294912 floats
constexpr int   NCOMP = B * C * P;       // 153600 complex
constexpr int   NROWS = B * C * H;       // 6144 image rows (= 384 * 16)
constexpr float EPSF  = 1.1920929e-07f;
constexpr float ATT_SCALE = 0.35355339059327373f;   // hd^-0.5
constexpr float TWO_PI    = 6.283185307179586f;

struct cplx { float x, y; };

__device__ __forceinline__ cplx cmul(cplx a, cplx b) {
  return { a.x*b.x - a.y*b.y, a.x*b.y + a.y*b.x };
}
__device__ __forceinline__ cplx cfma(cplx a, cplx b, cplx c) {
  return { fmaf(a.x, b.x, fmaf(-a.y, b.y, c.x)),
           fmaf(a.x, b.y, fmaf( a.y, b.x, c.y)) };
}

typedef float v2f __attribute__((ext_vector_type(2)));
typedef float v8f __attribute__((ext_vector_type(8)));

// wave32 f32 WMMA: D(16x16,f32) += A(16x4,f32) * B(4x16,f32)
__device__ __forceinline__ v8f wmma4(v2f a, v2f b, v8f c) {
  return __builtin_amdgcn_wmma_f32_16x16x4_f32(false, a, false, b, (short)0, c,
                                               false, false);
}

// =====================================================================
// Complex channel-mixing GEMM via V_WMMA_F32_16X16X4_F32 (wave32).
//   Y[b, o, p] = sum_i A[o,i] * X[b, i, p]  (+ bias[o]) (+ Res[b,o,p])
//   a_trans==0: A[o,i] = Wm[i*C + o]   ('io' einsum layout)
//   a_trans==1: A[o,i] = Wm[o*C + i]   (final_w: out = x @ fw.T)
// One wave per 16x16 complex tile. Complex mult = 4 real WMMAs:
//   Cr += Ar*Br + Ai*(-Bi); Ci += Ar*Bi + Ai*Br.
// Operand packing (ISA 7.12.2): lane half h=lane>>4, VGPR j of A/B holds
// K = 2*h + j; M/N = lane&15.  C/D: VGPR r holds M = r + 8*h, N = lane&15.
// =====================================================================
__global__ __launch_bounds__(32) void cgemm16_wmma(
    const cplx* __restrict__ Wm, const cplx* __restrict__ X,
    cplx* __restrict__ Y, const cplx* __restrict__ Res,
    const cplx* __restrict__ Bias, int a_trans)
{
  const int lane = threadIdx.x;
  const int hh = lane >> 4;
  const int ll = lane & 15;
  const int tn = blockIdx.x * 16;       // point tile   (75 tiles, exact)
  const int tm = blockIdx.y * 16;       // out-chan tile (4 tiles, exact)
  const int b  = blockIdx.z;
  const cplx* __restrict__ Xb = X + (size_t)b * C * P;

  v8f cr = {0.f,0.f,0.f,0.f,0.f,0.f,0.f,0.f};
  v8f ci = {0.f,0.f,0.f,0.f,0.f,0.f,0.f,0.f};

  for (int kb = 0; kb < C; kb += 4) {
    const int k0 = kb + 2*hh;
    cplx a0, a1;
    if (a_trans) { a0 = Wm[(tm+ll)*C + k0];  a1 = Wm[(tm+ll)*C + k0 + 1]; }
    else         { a0 = Wm[k0*C + tm+ll];    a1 = Wm[(k0+1)*C + tm+ll];   }
    const cplx b0 = Xb[(size_t)k0      * P + tn + ll];
    const cplx b1 = Xb[(size_t)(k0+1)  * P + tn + ll];
    if (kb + 4 < C)   // next K-slab of X -> global_prefetch_b8
      __builtin_prefetch(&Xb[(size_t)(kb + 4 + 2*hh) * P + tn + ll], 0, 1);

    const v2f are = { a0.x,  a1.x };
    const v2f aim = { a0.y,  a1.y };
    const v2f bre = { b0.x,  b1.x };
    const v2f bim = { b0.y,  b1.y };
    const v2f nbi = { -b0.y, -b1.y };

    cr = wmma4(are, bre, cr);
    cr = wmma4(aim, nbi, cr);
    ci = wmma4(are, bim, ci);
    ci = wmma4(aim, bre, ci);
  }

  const size_t colbase = (size_t)b * C * P + tn + ll;
  #pragma unroll
  for (int r = 0; r < 8; ++r) {
    const int row = tm + r + 8*hh;
    cplx val = { cr[r], ci[r] };
    if (Bias) { val.x += Bias[row].x; val.y += Bias[row].y; }
    const size_t off = (size_t)row * P + colbase;
    if (Res)  { val.x += Res[off].x;  val.y += Res[off].y; }
    Y[off] = val;
  }
}

// =====================================================================
// WMMA-based separable DFTs (48-point transforms as basis GEMMs).
// Bases are generated on-device into workspace; all matrix math runs on
// the f32 WMMA pipe instead of VALU sincos loops.
// =====================================================================

// rows r2c: out[row,kx] = sum_w x[row,w] * e^{-2pi i w kx/48}
// A = x rows (M=6144 rows, K=48), B = cos/sin basis [48][32] (kx padded to 32).
__global__ __launch_bounds__(32) void fft_rows_r2c_wmma(
    const float* __restrict__ X, cplx* __restrict__ out,
    const float* __restrict__ Bre, const float* __restrict__ Bim)
{
  const int lane = threadIdx.x, hh = lane >> 4, ll = lane & 15;
  const int tn = blockIdx.x * 16;     // kx tile (2 tiles; tile 1 part-masked)
  const int tm = blockIdx.y * 16;     // row tile (384, exact)
  v8f cr = {0.f,0.f,0.f,0.f,0.f,0.f,0.f,0.f};
  v8f ci = {0.f,0.f,0.f,0.f,0.f,0.f,0.f,0.f};
  for (int kb = 0; kb < W; kb += 4) {
    const int k0 = kb + 2*hh;
    const v2f a  = { X[(size_t)(tm+ll)*W + k0], X[(size_t)(tm+ll)*W + k0 + 1] };
    const v2f br = { Bre[k0*32 + tn + ll],      Bre[(k0+1)*32 + tn + ll] };
    const v2f bi = { Bim[k0*32 + tn + ll],      Bim[(k0+1)*32 + tn + ll] };
    cr = wmma4(a, br, cr);
    ci = wmma4(a, bi, ci);
  }
  const int kx = tn + ll;
  if (kx < WF) {
    #pragma unroll
    for (int r = 0; r < 8; ++r)
      out[(size_t)(tm + r + 8*hh) * WF + kx] = { cr[r], ci[r] };
  }
}

// cols c2c: out[bc,ky,kx] = sum_h E[ky,h] * in[bc,h,kx]  (complex x complex)
// Re/Im interleaved into K (K=96): B operand = raw complex pair at (h,kx)
// (single b64 load); the +/-Er/Ei recombination is baked into two A bases
// Are/Aim of shape [48][96].
__global__ __launch_bounds__(32) void fft_cols_c2c_wmma(
    const cplx* __restrict__ in, cplx* __restrict__ out,
    const float* __restrict__ Are, const float* __restrict__ Aim)
{
  const int lane = threadIdx.x, hh = lane >> 4, ll = lane & 15;
  const int tn = blockIdx.x * 16;     // kx tile (2 tiles; tile 1 part-masked)
  const int tm = blockIdx.y * 16;     // ky tile (3 tiles, exact)
  const int bc = blockIdx.z;          // 128 (b*C)
  const cplx* __restrict__ inb = in + (size_t)bc * H * WF;
  const int kx = min(tn + ll, WF - 1);   // clamp: garbage cols masked at store
  v8f cr = {0.f,0.f,0.f,0.f,0.f,0.f,0.f,0.f};
  v8f ci = {0.f,0.f,0.f,0.f,0.f,0.f,0.f,0.f};
  for (int kb = 0; kb < 2*H; kb += 4) {
    const int k0 = kb + 2*hh;           // even; k0 -> re, k0+1 -> im of h=k0/2
    const int h  = k0 >> 1;
    const cplx bv = inb[(size_t)h * WF + kx];
    const v2f bb = { bv.x, bv.y };
    const v2f ar = { Are[(tm+ll)*96 + k0], Are[(tm+ll)*96 + k0 + 1] };
    const v2f ai = { Aim[(tm+ll)*96 + k0], Aim[(tm+ll)*96 + k0 + 1] };
    cr = wmma4(ar, bb, cr);
    ci = wmma4(ai, bb, ci);
  }
  if (tn + ll < WF) {
    #pragma unroll
    for (int r = 0; r < 8; ++r)
      out[(size_t)bc * H * WF + (size_t)(tm + r + 8*hh) * WF + tn + ll] =
          { cr[r], ci[r] };
  }
}

// rows c2r (hermitian inverse along W):
//   out[row,w] = (1/48) * sum_{k=0..24} wgt_k*(Xr[k] cos - Xi[k] sin)
// A operand = raw interleaved 50-float complex row, K padded 50->52 with two
// zero basis rows (2-float overread stays inside workspace scratch t2).
__global__ __launch_bounds__(32) void fft_rows_c2r_wmma(
    const float* __restrict__ in, float* __restrict__ out,
    const float* __restrict__ Bb /*[52][48]*/)
{
  const int lane = threadIdx.x, hh = lane >> 4, ll = lane & 15;
  const int tn = blockIdx.x * 16;     // w tile (3 tiles, exact)
  const int tm = blockIdx.y * 16;     // row tile (384, exact)
  v8f cr = {0.f,0.f,0.f,0.f,0.f,0.f,0.f,0.f};
  for (int kb = 0; kb < 52; kb += 4) {
    const int k0 = kb + 2*hh;
    const v2f a = { in[(size_t)(tm+ll)*(2*WF) + k0],
                    in[(size_t)(tm+ll)*(2*WF) + k0 + 1] };
    const v2f b = { Bb[k0*W + tn + ll], Bb[(k0+1)*W + tn + ll] };
    cr = wmma4(a, b, cr);
  }
  #pragma unroll
  for (int r = 0; r < 8; ++r)
    out[(size_t)(tm + r + 8*hh) * W + tn + ll] = cr[r];
}

// ---------------- basis generation (tiny, per-call, deterministic) ----------
__global__ void init_basis_r2c(float* __restrict__ bre, float* __restrict__ bim)
{
  const int i = blockIdx.x * blockDim.x + threadIdx.x;   // 48*32
  if (i >= W * 32) return;
  const int w = i >> 5, kx = i & 31;
  float vr = 0.f, vi = 0.f;
  if (kx < WF) {
    float s, c;
    __sincosf((float)((w * kx) % W) * (-TWO_PI / W), &s, &c);
    vr = c; vi = s;
  }
  bre[i] = vr; bim[i] = vi;
}

__global__ void init_basis_col(float* __restrict__ fRe, float* __restrict__ fIm,
                               float* __restrict__ iRe, float* __restrict__ iIm)
{
  const int i = blockIdx.x * blockDim.x + threadIdx.x;   // 48*96
  if (i >= H * 96) return;
  const int ky = i / 96, kk = i % 96;
  const int h = kk >> 1;
  const bool odd = kk & 1;
  float s, c;
  __sincosf((float)((h * ky) % H) * (TWO_PI / H), &s, &c);
  const float fEr = c,              fEi = -s;              // exp(-i theta)
  const float iEr = c * (1.f/48.f), iEi =  s * (1.f/48.f); // exp(+i theta)/48
  // out_re row: [even]=Er, [odd]=-Ei ; out_im row: [even]=Ei, [odd]=Er
  fRe[i] = odd ? -fEi : fEr;
  fIm[i] = odd ?  fEr : fEi;
  iRe[i] = odd ? -iEi : iEr;
  iIm[i] = odd ?  iEr : iEi;
}

__global__ void init_basis_c2r(float* __restrict__ bb)
{
  const int i = blockIdx.x * blockDim.x + threadIdx.x;   // 52*48
  if (i >= 52 * W) return;
  const int kk = i / W, w = i % W;
  float v = 0.f;
  if (kk < 2 * WF) {
    const int k = kk >> 1;
    const float wgt = (k == 0 || k == W/2) ? 1.f : 2.f;  // hermitian fold
    float s, c;
    __sincosf((float)((w * k) % W) * (TWO_PI / W), &s, &c);
    v = ((kk & 1) ? (-wgt * s) : (wgt * c)) * (1.f / 48.f);
  }
  bb[i] = v;
}

// =====================================================================
// Complex RMSNorm over channels: one wave32 per spectral point.
//   ms = mean_c(z_c * z_c) + eps (complex);  y = z / csqrt(ms)
// =====================================================================
__global__ __launch_bounds__(32) void rmsnorm_c(const cplx* __restrict__ x,
                                                cplx* __restrict__ y)
{
  const int pt   = blockIdx.x;           // 0 .. B*P-1
  const int b    = pt / P;
  const int p    = pt % P;
  const int lane = threadIdx.x;
  const cplx* __restrict__ xb = x + (size_t)b * C * P + p;
  const cplx v0 = xb[(size_t)lane * P];
  const cplx v1 = xb[(size_t)(lane + 32) * P];
  cplx sq = cmul(v0, v0);
  { cplx t = cmul(v1, v1); sq.x += t.x; sq.y += t.y; }
  #pragma unroll
  for (int off = 16; off > 0; off >>= 1) {
    sq.x += __shfl_xor(sq.x, off, 32);
    sq.y += __shfl_xor(sq.y, off, 32);
  }
  const float msr = sq.x * (1.f / C) + EPSF;
  const float msi = sq.y * (1.f / C);
  const float mag = sqrtf(msr * msr + msi * msi);
  const float sr  = sqrtf(fmaxf(0.f, 0.5f * (mag + msr)));
  const float si  = copysignf(sqrtf(fmaxf(0.f, 0.5f * (mag - msr))), msi);
  const float inv = 1.f / mag;           // |csqrt(ms)|^2 == |ms| == mag
  const cplx d = { sr * inv, -si * inv };
  cplx* __restrict__ yb = y + (size_t)b * C * P + p;
  yb[(size_t)lane * P]        = cmul(v0, d);
  yb[(size_t)(lane + 32) * P] = cmul(v1, d);
}

// =====================================================================
// Fused causal complex attention, flash-style. One wave32 per output row x.
// =====================================================================
__global__ __launch_bounds__(32) void attention_row(
    const cplx* __restrict__ q, const cplx* __restrict__ k,
    const cplx* __restrict__ v, cplx* __restrict__ out)
{
  const int row = blockIdx.x;            // 0 .. B*NH*P-1
  const int x   = row % P;
  const int bn  = row / P;
  const int b   = bn / NH;
  const int n   = bn % NH;
  const int lane = threadIdx.x;
  const size_t hb = ((size_t)b * C + (size_t)n * HD) * P;
  const cplx* __restrict__ qb = q + hb;
  const cplx* __restrict__ kb = k + hb;
  const cplx* __restrict__ vb = v + hb;

  cplx qr[HD];
  #pragma unroll
  for (int i = 0; i < HD; ++i) qr[i] = qb[(size_t)i * P + x];

  float m = -1e30f, ssum = 0.f;          // finite sentinel avoids inf-inf NaNs
  cplx acc[HD];
  #pragma unroll
  for (int i = 0; i < HD; ++i) acc[i] = {0.f, 0.f};

  for (int y = lane; y <= x; y += 32) {
    cplx s = {0.f, 0.f};
    #pragma unroll
    for (int i = 0; i < HD; ++i) s = cfma(qr[i], kb[(size_t)i * P + y], s);
    s.x *= ATT_SCALE; s.y *= ATT_SCALE;
    const float mag = sqrtf(s.x * s.x + s.y * s.y);
    cplx ph = {1.f, 0.f};
    if (mag > 0.f) { const float im = 1.f / mag; ph = { s.x * im, s.y * im }; }
    const float nm   = fmaxf(m, mag);
    const float corr = expf(m - nm);     // 0 on first iteration (underflow)
    const float wgt  = expf(mag - nm);
    ssum = ssum * corr + wgt;
    #pragma unroll
    for (int i = 0; i < HD; ++i) {
      const cplx vv = vb[(size_t)i * P + y];
      const cplx wv = { wgt * (ph.x * vv.x - ph.y * vv.y),
                        wgt * (ph.x * vv.y + ph.y * vv.x) };
      acc[i].x = acc[i].x * corr + wv.x;
      acc[i].y = acc[i].y * corr + wv.y;
    }
    m = nm;
  }

  #pragma unroll
  for (int off = 16; off > 0; off >>= 1) {
    const float m2 = __shfl_xor(m, off, 32);
    const float s2 = __shfl_xor(ssum, off, 32);
    const float nm = fmaxf(m, m2);
    const float f1 = expf(m - nm);
    const float f2 = expf(m2 - nm);
    ssum = ssum * f1 + s2 * f2;
    #pragma unroll
    for (int i = 0; i < HD; ++i) {
      const float ax = __shfl_xor(acc[i].x, off, 32);
      const float ay = __shfl_xor(acc[i].y, off, 32);
      acc[i].x = acc[i].x * f1 + ax * f2;
      acc[i].y = acc[i].y * f1 + ay * f2;
    }
    m = nm;
  }

  if (lane == 0) {
    const float inv = 1.f / ssum;
    cplx* __restrict__ ob = out + hb;
    #pragma unroll
    for (int i = 0; i < HD; ++i)
      ob[(size_t)i * P + x] = { acc[i].x * inv, acc[i].y * inv };
  }
}

// ---------------- elementwise ----------------
__global__ void silu_kernel(float* __restrict__ x, int n)
{
  const int i = blockIdx.x * blockDim.x + threadIdx.x;
  if (i < n) { const float t = x[i]; x[i] = t / (1.f + expf(-t)); }
}

__global__ void add_kernel(float* __restrict__ a, const float* __restrict__ bsrc, int n)
{
  const int i = blockIdx.x * blockDim.x + threadIdx.x;
  if (i < n) a[i] += bsrc[i];
}

// =====================================================================
// Host-side orchestration (graph-capture safe: only async launches).
// =====================================================================
extern "C" void kernel_launch(void* const* d_in, const int* in_sizes, int n_in,
                              void* d_out, int out_size, void* d_ws, size_t ws_size,
                              hipStream_t stream)
{
  (void)in_sizes; (void)n_in; (void)out_size; (void)ws_size;

  const float* x   = (const float*)d_in[0];
  const cplx*  w1  = (const cplx*)d_in[1];
  const cplx*  wq  = (const cplx*)d_in[2];
  const cplx*  wk  = (const cplx*)d_in[3];
  const cplx*  wv  = (const cplx*)d_in[4];
  const cplx*  wo  = (const cplx*)d_in[5];
  const cplx*  wff = (const cplx*)d_in[6];
  const cplx*  fw  = (const cplx*)d_in[7];
  const cplx*  fb  = (const cplx*)d_in[8];

  float* outr = (float*)d_out;                       // [2,64,48,48] real
  cplx*  yout = (cplx*)((float*)d_out + NREAL);      // [2,64,48,25] complex (output #2)

  float* ws = (float*)d_ws;
  const size_t CF = (size_t)NCOMP * 2;               // floats per complex field
  cplx* bufA = (cplx*)(ws + 0 * CF);
  cplx* bufB = (cplx*)(ws + 1 * CF);
  cplx* t1   = (cplx*)(ws + 2 * CF);
  cplx* t2   = (cplx*)(ws + 3 * CF);                 // inv_fft scratch (interior!)
  cplx* xn   = (cplx*)(ws + 4 * CF);
  cplx* qb   = (cplx*)(ws + 5 * CF);
  cplx* kb   = (cplx*)(ws + 6 * CF);
  cplx* vb   = (cplx*)(ws + 7 * CF);
  cplx* ab   = (cplx*)(ws + 8 * CF);
  float* tr  = (float*)(ws + 9 * CF);                // real scratch [2,64,48,48]

  size_t boff = 9 * CF + (size_t)NREAL;
  float* basR2Cre = ws + boff; boff += (size_t)W * 32;
  float* basR2Cim = ws + boff; boff += (size_t)W * 32;
  float* basColFre = ws + boff; boff += (size_t)H * 96;
  float* basColFim = ws + boff; boff += (size_t)H * 96;
  float* basColIre = ws + boff; boff += (size_t)H * 96;
  float* basColIim = ws + boff; boff += (size_t)H * 96;
  float* basC2R    = ws + boff; boff += (size_t)52 * W;

  const dim3 bl(256), bW(32);
  const dim3 gReal((NREAL + 255) / 256);
  const dim3 gGemm(P / 16, C / 16, B);               // 75 x 4 x 2, wave32 blocks
  const dim3 gNorm(B * P);
  const dim3 gAttn(B * NH * P);
  const dim3 gRowsF(2, NROWS / 16);                  // r2c:   2 kx-tiles x 384
  const dim3 gCols(2, H / 16, B * C);                // c2c:   2 x 3 x 128
  const dim3 gRowsI(W / 16, NROWS / 16);             // c2r:   3 w-tiles x 384

  // generate DFT bases (tiny; runs every call, deterministic)
  init_basis_r2c<<<dim3((W*32 + 255)/256),  bl, 0, stream>>>(basR2Cre, basR2Cim);
  init_basis_col<<<dim3((H*96 + 255)/256),  bl, 0, stream>>>(basColFre, basColFim,
                                                             basColIre, basColIim);
  init_basis_c2r<<<dim3((52*W + 255)/256),  bl, 0, stream>>>(basC2R);

  auto fwd_fft = [&](const float* rin, cplx* scratch, cplx* cout) {
    fft_rows_r2c_wmma<<<gRowsF, bW, 0, stream>>>(rin, scratch, basR2Cre, basR2Cim);
    fft_cols_c2c_wmma<<<gCols, bW, 0, stream>>>(scratch, cout, basColFre, basColFim);
  };
  auto inv_fft = [&](const cplx* cin, float* rout) {   // scratch is always t2
    fft_cols_c2c_wmma<<<gCols, bW, 0, stream>>>(cin, t2, basColIre, basColIim);
    fft_rows_c2r_wmma<<<gRowsI, bW, 0, stream>>>((const float*)t2, rout, basC2R);
  };

  // xft = rfft2(x)
  fwd_fft(x, t1, bufA);
  cplx* cur = bufA;
  cplx* alt = bufB;

  for (int l = 0; l < 2; ++l) {
    const cplx* W1 = w1  + (size_t)l * C * C;
    const cplx* Wq = wq  + (size_t)l * C * C;
    const cplx* Wk = wk  + (size_t)l * C * C;
    const cplx* Wv = wv  + (size_t)l * C * C;
    const cplx* Wo = wo  + (size_t)l * C * C;
    const cplx* Wf = wff + (size_t)l * C * C;

    // spectral conv + silu re-projection: xft = rfft2(silu(irfft2(w1 . xft)))
    cgemm16_wmma<<<gGemm, bW, 0, stream>>>(W1, cur, t1, nullptr, nullptr, 0);
    inv_fft(t1, tr);
    silu_kernel<<<gReal, bl, 0, stream>>>(tr, NREAL);
    fwd_fft(tr, t1, cur);

    // transformer block (pre-norm residual)
    rmsnorm_c<<<gNorm, bW, 0, stream>>>(cur, xn);
    cgemm16_wmma<<<gGemm, bW, 0, stream>>>(Wq, xn, qb, nullptr, nullptr, 0);
    cgemm16_wmma<<<gGemm, bW, 0, stream>>>(Wk, xn, kb, nullptr, nullptr, 0);
    cgemm16_wmma<<<gGemm, bW, 0, stream>>>(Wv, xn, vb, nullptr, nullptr, 0);
    attention_row<<<gAttn, bW, 0, stream>>>(qb, kb, vb, ab);
    cgemm16_wmma<<<gGemm, bW, 0, stream>>>(Wo, ab, alt, cur, nullptr, 0);
    { cplx* tmp = cur; cur = alt; alt = tmp; }

    // feed-forward: xft += rfft2(silu(irfft2(wff . rmsnorm(xft))))
    rmsnorm_c<<<gNorm, bW, 0, stream>>>(cur, xn);
    cgemm16_wmma<<<gGemm, bW, 0, stream>>>(Wf, xn, t1, nullptr, nullptr, 0);
    inv_fft(t1, tr);
    silu_kernel<<<gReal, bl, 0, stream>>>(tr, NREAL);
    fwd_fft(tr, t2, t1);
    add_kernel<<<dim3((NCOMP * 2 + 255) / 256), bl, 0, stream>>>(
        (float*)cur, (const float*)t1, NCOMP * 2);
  }

  // final complex linear (xl @ fw.T + fb) -> second output, then irfft2 -> first
  cgemm16_wmma<<<gGemm, bW, 0, stream>>>(fw, cur, yout, nullptr, fb, 1);
  inv_fft(yout, outr);
}